// virtualFP_38800734552479
// MI455X (gfx1250) — compile-verified
//
#include <hip/hip_runtime.h>

typedef float v2f __attribute__((ext_vector_type(2)));
typedef float v8f __attribute__((ext_vector_type(8)));

constexpr int M_DIM = 8192;
constexpr int NR    = 8192;
constexpr int TILES_PER_WAVE  = 4;                                   // 4 16x16 tiles per wave
constexpr int WAVES_PER_BLOCK = 8;                                   // 256 threads, wave32
constexpr int BLOCK_COLS      = WAVES_PER_BLOCK * TILES_PER_WAVE * 16; // 512 cols per block

// out[m][n] = C[m][n] * D[n], computed as 16x16 tiles:
//   out_tile = sum_{j=0..3} A_j(16x4) x B_j(4x16)   with B_j = diagonal slice of D.
// Pure streaming kernel: NT loads of C, NT stores of out (512MB working set >> 192MB L2,
// zero reuse), WMMA as the exact fp32 column-scaling engine.
__global__ __launch_bounds__(256) void colscale_diag_wmma(const float* __restrict__ C,
                                                          const float* __restrict__ D,
                                                          float* __restrict__ out) {
    const int lane = threadIdx.x & 31;
    const int wave = threadIdx.x >> 5;
    const int n = lane & 15;   // column-within-tile (B/C/D operand layouts)
    const int h = lane >> 4;   // half-wave select

    const int row0 = blockIdx.y * 16;
    const int colw = blockIdx.x * BLOCK_COLS + wave * (TILES_PER_WAVE * 16);

    // A operand: this lane always reads row row0 + n (both halves; halves differ in K)
    const size_t rowA_off = (size_t)(row0 + n) * NR;

#pragma unroll
    for (int t = 0; t < TILES_PER_WAVE; ++t) {
        const int col0 = colw + 16 * t;

        // Diagonal value for this lane's output column (tiny, hot: default/RT caching)
        const float dval = D[col0 + n];

        v8f acc = {0.f, 0.f, 0.f, 0.f, 0.f, 0.f, 0.f, 0.f};

#pragma unroll
        for (int j = 0; j < 4; ++j) {
            // A chunk j: lane loads C[row0+n][col0 + 4j + 2h .. +1]  (8B NT load)
            const v2f a = __builtin_nontemporal_load(
                reinterpret_cast<const v2f*>(C + rowA_off + (size_t)(col0 + 4 * j + 2 * h)));

            // B chunk j (4x16): b.x = B[2h][n], b.y = B[2h+1][n];
            // nonzero only on the diagonal of this K-chunk.
            v2f b;
            b.x = (n == 4 * j + 2 * h)     ? dval : 0.0f;
            b.y = (n == 4 * j + 2 * h + 1) ? dval : 0.0f;

            // acc = A_j x B_j + acc   (v_wmma_f32_16x16x4_f32)
            acc = __builtin_amdgcn_wmma_f32_16x16x4_f32(
                false, a, false, b, (short)0, acc, false, false);
        }

        // C/D layout: VGPR v holds out[row0 + v + 8h][col0 + n]; NT stores,
        // 16 lanes per half-wave coalesce to contiguous 64B row segments.
#pragma unroll
        for (int v = 0; v < 8; ++v) {
            __builtin_nontemporal_store(
                acc[v], out + (size_t)(row0 + v + 8 * h) * NR + (size_t)(col0 + n));
        }
    }
}

extern "C" void kernel_launch(void* const* d_in, const int* in_sizes, int n_in,
                              void* d_out, int out_size, void* d_ws, size_t ws_size,
                              hipStream_t stream) {
    const float* C = (const float*)d_in[0];   // [M, NR] fp32
    const float* D = (const float*)d_in[1];   // [NR]    fp32
    float* out = (float*)d_out;               // [M, NR] fp32

    dim3 grid(NR / BLOCK_COLS, M_DIM / 16);   // (16, 512)
    colscale_diag_wmma<<<grid, WAVES_PER_BLOCK * 32, 0, stream>>>(C, D, out);
}